// MultiheadAttentionZCross_58145267253824
// MI455X (gfx1250) — compile-verified
//
#include <hip/hip_runtime.h>

typedef __attribute__((ext_vector_type(2))) float v2f;
typedef __attribute__((ext_vector_type(8))) float v8f;

#define T_LEN 2048
#define S_LEN 2048
#define BSZ   8
#define EDIM  512
#define SCALE 0.125f          // (512/8)^-0.5
#define NEG_BIG (-3.0e38f)

// ---------------------------------------------------------------------------
// Kernel A: C[t,b,:] = softmax_s( (q1*SCALE) . k  + mask ) @ K     -> d_out
// Block = 256 threads (8 waves). Waves pair up: tile = wave>>1 owns 16 t-rows,
// eh = wave&1 owns a 256-wide e-half (acc[16] = 128 VGPRs, fully unrolled so
// the array is register-promoted -- no scratch).
// LDS (floats):
//   Qa   [4][128][32][2]  A-operand layout of scaled Q  (32768 f, per t-tile)
//   Ksc  [128][32][2]     B-operand layout of K tile for Q@K^T (8192 f)
//   Kval [32][4][32][2]   B-operand layout of K tile for P@K   (8192 f)
//   Pt   [8][16*20]       per-wave P transpose scratch (2560 f)
// total 51712 floats = 206848 B dynamic LDS
// ---------------------------------------------------------------------------
__global__ __launch_bounds__(256, 1)
void attn_kernel(const float* __restrict__ q1,
                 const float* __restrict__ kmat,
                 const int*   __restrict__ mask,
                 float*       __restrict__ outC) {
  extern __shared__ float smem[];
  float* Qa   = smem;            // 32768
  float* Ksc  = smem + 32768;    // 8192
  float* Kval = smem + 40960;    // 8192
  float* Pt   = smem + 49152;    // 2560

  const int tid  = threadIdx.x;      // 0..255
  const int wave = tid >> 5;         // 0..7
  const int lane = tid & 31;
  const int hf   = lane >> 4;        // lane half
  const int lc   = lane & 15;
  const int tile = wave >> 1;        // t-tile 0..3
  const int eh   = wave & 1;         // e-half 0/1
  const int b    = blockIdx.y;
  const int t0   = blockIdx.x * 64;
  const int tw   = t0 + tile * 16;

  // ---- stage scaled Q block (64x512) into per-tile WMMA-A layout ----
  for (int it = 0; it < 32; ++it) {
    int fidx = it * 256 + tid;           // float4 index in 64x128
    int mrow = fidx >> 7;                // 0..63
    int e4   = (fidx & 127) << 2;
    float4 v = *(const float4*)(q1 + (((size_t)(t0 + mrow) * BSZ + b) * EDIM + e4));
    int w = mrow >> 4, m16 = mrow & 15, kk = e4 >> 2;
    float* bp = Qa + ((w * 128 + kk) * 32) * 2;
    bp[m16 * 2 + 0]        = v.x * SCALE;
    bp[m16 * 2 + 1]        = v.y * SCALE;
    bp[(m16 + 16) * 2 + 0] = v.z * SCALE;
    bp[(m16 + 16) * 2 + 1] = v.w * SCALE;
  }

  v8f zero = {};
  v8f acc[16];                         // 16 e-tiles of this wave's half
#pragma unroll
  for (int j = 0; j < 16; ++j) acc[j] = zero;
  float Mrow[8], Lrow[8];
#pragma unroll
  for (int r = 0; r < 8; ++r) { Mrow[r] = NEG_BIG; Lrow[r] = 0.f; }

  const v2f* qa = (const v2f*)(Qa + (size_t)tile * 8192) + lane;

  for (int s0 = 0; s0 < S_LEN; s0 += 16) {
    __syncthreads();
    // ---- stage K tile (16x512) in both operand layouts ----
    for (int it = 0; it < 8; ++it) {
      int fidx = it * 256 + tid;         // 0..2047
      int j  = fidx >> 7;                // s row 0..15
      int e4 = (fidx & 127) << 2;
      float4 v = *(const float4*)(kmat + (((size_t)(s0 + j) * BSZ + b) * EDIM + e4));
      int kk = e4 >> 2;
      float* kb = Ksc + (kk * 32) * 2;   // B for Q@K^T: lane=n(s), k=e
      kb[j * 2 + 0]        = v.x;
      kb[j * 2 + 1]        = v.y;
      kb[(j + 16) * 2 + 0] = v.z;
      kb[(j + 16) * 2 + 1] = v.w;
      // B for P@K: lane=n(e col), k=s
      int st = j >> 2, rr = j & 3, h2 = rr >> 1, cm = rr & 1;
      int jt = e4 >> 4, c0 = e4 & 15;
      float* vb = Kval + (((jt * 4 + st) * 32) + 16 * h2) * 2 + cm;
      vb[(c0 + 0) * 2] = v.x;
      vb[(c0 + 1) * 2] = v.y;
      vb[(c0 + 2) * 2] = v.z;
      vb[(c0 + 3) * 2] = v.w;
    }
    __syncthreads();

    if (s0 + 16 < S_LEN) {  // touch next tile (global_prefetch)
      const float* np = kmat + (((size_t)(s0 + 16 + (tid >> 4)) * BSZ + b) * EDIM + (tid & 15) * 32);
      __builtin_prefetch(np, 0, 1);
    }

    // ---- S = Q_tile @ K_tile^T  (16x16, contraction over e=512) ----
    v8f s = zero;
    const v2f* kb2 = (const v2f*)Ksc + lane;
#pragma unroll 8
    for (int kk = 0; kk < 128; ++kk) {
      s = __builtin_amdgcn_wmma_f32_16x16x4_f32(false, qa[kk * 32],
                                                false, kb2[kk * 32],
                                                (short)0, s, false, false);
    }

    // ---- mask + online softmax (redundant across the wave pair) ----
    const bool masked = (mask[b * S_LEN + s0 + lc] != 0);
    if (masked) {
#pragma unroll
      for (int r = 0; r < 8; ++r) s[r] = NEG_BIG;
    }
    float alpha[8];
#pragma unroll
    for (int r = 0; r < 8; ++r) {
      float v = s[r];
      v = fmaxf(v, __shfl_xor(v, 1));
      v = fmaxf(v, __shfl_xor(v, 2));
      v = fmaxf(v, __shfl_xor(v, 4));
      v = fmaxf(v, __shfl_xor(v, 8));       // row max within 16-lane half
      float mnew = fmaxf(Mrow[r], v);
      alpha[r]   = __expf(Mrow[r] - mnew);
      Mrow[r]    = mnew;
    }
#pragma unroll
    for (int r = 0; r < 8; ++r) {
      float p = masked ? 0.f : __expf(s[r] - Mrow[r]);
      s[r] = p;
      float v = p;
      v += __shfl_xor(v, 1);
      v += __shfl_xor(v, 2);
      v += __shfl_xor(v, 4);
      v += __shfl_xor(v, 8);
      Lrow[r] = Lrow[r] * alpha[r] + v;
    }
    v8f av;
#pragma unroll
    for (int r = 0; r < 8; ++r) av[r] = alpha[r];
#pragma unroll
    for (int j = 0; j < 16; ++j) acc[j] *= av;

    // ---- transpose P (D-layout -> A-layout) via per-wave LDS scratch ----
    float* ptw = Pt + wave * 320;           // row stride 20 (bank-friendly)
#pragma unroll
    for (int r = 0; r < 8; ++r) ptw[(r + 8 * hf) * 20 + lc] = s[r];
    v2f pa[4];
#pragma unroll
    for (int st = 0; st < 4; ++st)
      pa[st] = *(const v2f*)(ptw + lc * 20 + 4 * st + 2 * hf);

    // ---- C_half += P @ K_tile_half  (16x256, contraction over s=16) ----
    // FULLY unrolled: constant indices keep acc[] register-promoted.
    const v2f* kvb = (const v2f*)Kval + (size_t)(eh * 16) * 4 * 32 + lane;
#pragma unroll
    for (int jt = 0; jt < 16; ++jt) {
      v8f c = acc[jt];
#pragma unroll
      for (int st = 0; st < 4; ++st) {
        c = __builtin_amdgcn_wmma_f32_16x16x4_f32(false, pa[st],
                                                  false, kvb[(jt * 4 + st) * 32],
                                                  (short)0, c, false, false);
      }
      acc[jt] = c;
    }
  }

  // ---- normalize by row sums and store this wave's e-half into d_out ----
  v8f rv;
#pragma unroll
  for (int r = 0; r < 8; ++r) rv[r] = 1.0f / Lrow[r];
#pragma unroll
  for (int jt = 0; jt < 16; ++jt) {
    v8f c = acc[jt] * rv;
    int col = 16 * (eh * 16 + jt) + lc;
#pragma unroll
    for (int r = 0; r < 8; ++r)
      outC[((size_t)(tw + r + 8 * hf) * BSZ + b) * EDIM + col] = c[r];
  }
}

// ---------------------------------------------------------------------------
// Kernel B (in place on d_out): out_row = C_row @ W^T + bias
// Block = 256 threads (8 waves), 128 rows per WG (rows = flattened t*8+b).
// LDS: Ca [8][128][32][2] = 65536 f (staged BEFORE overwrite), Wb [128][32][2]
// total 73728 floats = 294912 B dynamic LDS
// ---------------------------------------------------------------------------
__global__ __launch_bounds__(256, 1)
void proj_kernel(const float* __restrict__ W,
                 const float* __restrict__ bias,
                 float*       __restrict__ io) {
  extern __shared__ float smem[];
  float* Ca = smem;            // 65536
  float* Wb = smem + 65536;    // 8192

  const int tid  = threadIdx.x;    // 0..255
  const int wave = tid >> 5;
  const int lane = tid & 31;
  const int hf   = lane >> 4;
  const int lc   = lane & 15;
  const size_t row0 = (size_t)blockIdx.x * 128;

  // stage C rows into per-wave A layout (then rows may be overwritten)
  for (int it = 0; it < 64; ++it) {
    int fidx = it * 256 + tid;          // 0..16383 float4
    int mrow = fidx >> 7;               // 0..127
    int e4   = (fidx & 127) << 2;
    float4 v = *(const float4*)(io + ((row0 + mrow) * EDIM + e4));
    int w = mrow >> 4, m16 = mrow & 15, kk = e4 >> 2;
    float* bp = Ca + (((size_t)w * 128 + kk) * 32) * 2;
    bp[m16 * 2 + 0]        = v.x;
    bp[m16 * 2 + 1]        = v.y;
    bp[(m16 + 16) * 2 + 0] = v.z;
    bp[(m16 + 16) * 2 + 1] = v.w;
  }
  __syncthreads();

  const v2f* ca = (const v2f*)(Ca + (size_t)wave * 8192) + lane;
  for (int jt = 0; jt < 32; ++jt) {
    __syncthreads();                    // protect Wb reuse
    // stage W rows f = 16*jt .. +15 (all 512 e) in B-operand layout
    for (int it = 0; it < 8; ++it) {
      int fidx = it * 256 + tid;        // 0..2047 float4
      int fr = fidx >> 7;               // 0..15
      int e4 = (fidx & 127) << 2;
      float4 v = *(const float4*)(W + (((size_t)(16 * jt + fr)) * EDIM + e4));
      int kk = e4 >> 2;
      float* wb = Wb + (kk * 32) * 2;
      wb[fr * 2 + 0]        = v.x;
      wb[fr * 2 + 1]        = v.y;
      wb[(fr + 16) * 2 + 0] = v.z;
      wb[(fr + 16) * 2 + 1] = v.w;
    }
    __syncthreads();

    v8f c = {};
    const v2f* wbv = (const v2f*)Wb + lane;
#pragma unroll 8
    for (int kk = 0; kk < 128; ++kk) {
      c = __builtin_amdgcn_wmma_f32_16x16x4_f32(false, ca[kk * 32],
                                                false, wbv[kk * 32],
                                                (short)0, c, false, false);
    }
    float bf = bias[16 * jt + lc];
#pragma unroll
    for (int r = 0; r < 8; ++r)
      io[(row0 + wave * 16 + r + 8 * hf) * EDIM + 16 * jt + lc] = c[r] + bf;
  }
}

extern "C" void kernel_launch(void* const* d_in, const int* in_sizes, int n_in,
                              void* d_out, int out_size, void* d_ws, size_t ws_size,
                              hipStream_t stream) {
  const float* q1   = (const float*)d_in[0];
  const float* kmat = (const float*)d_in[1];
  const int*   mask = (const int*)d_in[2];
  const float* W    = (const float*)d_in[3];
  const float* bias = (const float*)d_in[4];
  float* out = (float*)d_out;

  dim3 gA(T_LEN / 64, BSZ);
  attn_kernel<<<gA, dim3(256), 51712 * sizeof(float), stream>>>(q1, kmat, mask, out);

  dim3 gB((T_LEN * BSZ) / 128);
  proj_kernel<<<gB, dim3(256), 73728 * sizeof(float), stream>>>(W, bias, out);
}